// DeepFloorFusedMixer_9947144257889
// MI455X (gfx1250) — compile-verified
//
#include <hip/hip_runtime.h>
#include <hip/hip_bf16.h>

typedef float v8f __attribute__((ext_vector_type(8)));
typedef float v2f __attribute__((ext_vector_type(2)));
typedef unsigned int u32x4 __attribute__((ext_vector_type(4)));
typedef int i32x8 __attribute__((ext_vector_type(8)));
typedef int i32x4 __attribute__((ext_vector_type(4)));

#define Dm 128
#define Bm 4
#define Tm 1024
#define BT 4096        // Bm*Tm
#define DD 16384       // Dm*Dm
#define NSEG 16
#define SEGLEN 64
#define QHALF 7.0f     // (15-1)/2

// workspace float offsets
#define OFF_QWT 0
#define OFF_KWT (1*DD)
#define OFF_VWT (2*DD)
#define OFF_SWT (3*DD)
#define OFF_OWT (4*DD)
#define OFF_TW  (5*DD)
#define OFF_MAX (6*DD)
#define OFF_Q   (6*DD + 16)
#define OFF_KS  (OFF_Q  + BT*Dm)
#define OFF_VS  (OFF_KS + BT*Dm)
#define OFF_MIX (OFF_VS + BT*Dm)
#define OFF_SEG (OFF_MIX + BT*Dm)   // Bm*8*NSEG*2048 = 1048576 floats

__device__ __forceinline__ unsigned rnd32(unsigned long long x) {
    x += 0x9E3779B97F4A7C15ull;
    x = (x ^ (x >> 30)) * 0xBF58476D1CE4E5B9ull;
    x = (x ^ (x >> 27)) * 0x94D049BB133111EBull;
    x ^= x >> 31;
    return (unsigned)(x >> 32);
}

__device__ __forceinline__ float squant(float x, unsigned long long idx,
                                        float step, float invstep, bool quant) {
    if (!quant) return x;
    float s = fminf(fmaxf(x * invstep, -QHALF), QHALF);
    float lo = floorf(s);
    float u = (float)rnd32(idx) * 2.3283064365386963e-10f;  // [0,1)
    float r = (u < (s - lo)) ? (lo + 1.0f) : lo;
    return r * step;
}

// ---------------------------------------------------------------------------
// K1: spectral clamps of the 5 projection weights + HiPPO transition (1 block)
// ---------------------------------------------------------------------------
__device__ void blk_sum(float* red, int tid) {
    __syncthreads();
    for (int s = 128; s > 0; s >>= 1) {
        if (tid < s) red[tid] += red[tid + s];
        __syncthreads();
    }
}

__global__ void __launch_bounds__(256)
prep_weights(const float* qW, const float* kW, const float* vW,
             const float* sW, const float* oW,
             const float* hl, const float* hr, float* ws) {
    const int tid = threadIdx.x;
    __shared__ float sv[Dm], su[Dm], red[256], sscale;
    float* tw = ws + OFF_TW;

    // build HiPPO-LegS + low-rank delta into tw slot
    for (int e = tid; e < DD; e += 256) {
        int i = e >> 7, j = e & 127;
        float base = 0.0f;
        if (j < i)       base = -sqrtf(2.0f * i + 1.0f) * sqrtf(2.0f * j + 1.0f);
        else if (j == i) base = -((float)i + 1.0f);
        float dot = 0.0f;
        #pragma unroll
        for (int r = 0; r < 16; r++) dot += hl[i * 16 + r] * hr[j * 16 + r];
        tw[e] = base + 0.01f * dot;
    }
    if (tid == 0) ws[OFF_MAX] = 0.0f;
    __threadfence();
    __syncthreads();

    const float proj_gain = 0.9486832980505138f;  // sqrt(0.9)
    for (int m = 0; m < 6; m++) {
        const float* M;
        float target;
        switch (m) {
            case 0: M = qW; target = proj_gain; break;
            case 1: M = kW; target = proj_gain; break;
            case 2: M = vW; target = proj_gain; break;
            case 3: M = sW; target = 1.0f;      break;
            case 4: M = oW; target = proj_gain; break;
            default: M = tw; target = 0.95f;    break;
        }
        if (tid < Dm) sv[tid] = 0.08838834764831845f;  // 1/sqrt(128)
        __syncthreads();

        for (int pit = 0; pit < 4; pit++) {
            if (tid < Dm) {
                float a = 0.0f;
                for (int j = 0; j < Dm; j++) a += M[tid * Dm + j] * sv[j];
                su[tid] = a;
            }
            red[tid] = (tid < Dm) ? su[tid] * su[tid] : 0.0f;
            blk_sum(red, tid);
            {
                float dn = fmaxf(sqrtf(red[0]), 1e-12f);
                if (tid < Dm) su[tid] /= dn;
            }
            __syncthreads();
            if (tid < Dm) {
                float a = 0.0f;
                for (int i = 0; i < Dm; i++) a += M[i * Dm + tid] * su[i];
                sv[tid] = a;
            }
            red[tid] = (tid < Dm) ? sv[tid] * sv[tid] : 0.0f;
            blk_sum(red, tid);
            {
                float dn = fmaxf(sqrtf(red[0]), 1e-12f);
                if (tid < Dm) sv[tid] /= dn;
            }
            __syncthreads();
        }
        float part = 0.0f;
        if (tid < Dm) {
            float a = 0.0f;
            for (int j = 0; j < Dm; j++) a += M[tid * Dm + j] * sv[j];
            part = su[tid] * a;
        }
        red[tid] = (tid < Dm) ? part : 0.0f;
        blk_sum(red, tid);
        if (tid == 0) {
            float sg = fmaxf(fabsf(red[0]), 1e-6f);
            sscale = fminf(target / sg, 1.0f);
        }
        __syncthreads();
        float scl = sscale;
        if (m < 5) {
            float* dstT = ws + m * DD;  // transposed: dstT[d*128+o] = W[o][d]*scl
            for (int e = tid; e < DD; e += 256) {
                int o = e >> 7, d = e & 127;
                dstT[d * Dm + o] = M[e] * scl;
            }
        } else {
            for (int e = tid; e < DD; e += 256) tw[e] *= scl;
        }
        __threadfence();
        __syncthreads();
    }
}

// ---------------------------------------------------------------------------
// K2: per-token rmsnorm + q/k/v/select projections, ks/vs, global max reduce
// ---------------------------------------------------------------------------
__global__ void __launch_bounds__(128)
project_tokens(const float* state, float* ws) {
    const int tok = blockIdx.x;
    const int d = threadIdx.x;
    __shared__ float sn[Dm], red[Dm], mk_s;
    const float* qwT = ws + OFF_QWT;
    const float* kwT = ws + OFF_KWT;
    const float* vwT = ws + OFF_VWT;
    const float* swT = ws + OFF_SWT;

    float x = state[tok * Dm + d];
    red[d] = x * x;
    __syncthreads();
    for (int s = 64; s > 0; s >>= 1) {
        if (d < s) red[d] += red[d + s];
        __syncthreads();
    }
    float rn = rsqrtf(red[0] * (1.0f / Dm) + 1e-6f);
    sn[d] = x * rn;
    __syncthreads();

    float aq = 0.0f, ak = 0.0f, av = 0.0f, as = 0.0f;
    for (int j = 0; j < Dm; j++) {
        float n = sn[j];
        aq += n * qwT[j * Dm + d];
        ak += n * kwT[j * Dm + d];
        av += n * vwT[j * Dm + d];
        as += n * swT[j * Dm + d];
    }
    float qv = (aq > 0.0f) ? (aq + 1.0f) : expf(aq);  // elu(x)+1
    float kv = (ak > 0.0f) ? (ak + 1.0f) : expf(ak);
    float sel = 1.0f / (1.0f + expf(-as));
    float ksv = kv * sel, vsv = av * sel;

    ws[OFF_Q  + tok * Dm + d] = qv;
    ws[OFF_KS + tok * Dm + d] = ksv;
    ws[OFF_VS + tok * Dm + d] = vsv;

    __syncthreads();
    red[d] = fabsf(ksv);
    __syncthreads();
    for (int s = 64; s > 0; s >>= 1) {
        if (d < s) red[d] = fmaxf(red[d], red[d + s]);
        __syncthreads();
    }
    if (d == 0) mk_s = red[0];
    __syncthreads();
    red[d] = fabsf(vsv);
    __syncthreads();
    for (int s = 64; s > 0; s >>= 1) {
        if (d < s) red[d] = fmaxf(red[d], red[d + s]);
        __syncthreads();
    }
    if (d == 0)
        atomicMax((int*)(ws + OFF_MAX), __float_as_int(mk_s * red[0]));
}

// ---------------------------------------------------------------------------
// K3a: per-segment sums of quantized outer products  (b,kt,g) blocks
// ---------------------------------------------------------------------------
__global__ void __launch_bounds__(256)
seg_outer(float* ws) {
    const int g = blockIdx.x, kt = blockIdx.y, b = blockIdx.z;
    const int tid = threadIdx.x;
    const int k = tid & 15, i0 = tid >> 4;  // i0: 0..15
    const float* ks = ws + OFF_KS;
    const float* vs = ws + OFF_VS;
    float maxabs = ws[OFF_MAX];
    float step = maxabs / QHALF;
    bool quant = maxabs > 1e-12f;
    float inv = quant ? 1.0f / step : 0.0f;

    __shared__ float ks_s[Dm], vs_s[16];
    float acc[8];
    #pragma unroll
    for (int m = 0; m < 8; m++) acc[m] = 0.0f;

    for (int tt = 0; tt < SEGLEN; tt++) {
        int t = g * SEGLEN + tt;
        long long base = (long long)(b * Tm + t) * Dm;
        if (tid < Dm)            ks_s[tid] = ks[base + tid];
        else if (tid < Dm + 16)  vs_s[tid - Dm] = vs[base + kt * 16 + (tid - Dm)];
        __syncthreads();
        float vv = vs_s[k];
        int jglob = kt * 16 + k;
        #pragma unroll
        for (int m = 0; m < 8; m++) {
            int i = i0 + 16 * m;
            unsigned long long idx =
                (((unsigned long long)(b * Tm + t) * Dm + i) * Dm + jglob);
            acc[m] += squant(ks_s[i] * vv, idx, step, inv, quant);
        }
        __syncthreads();
    }
    float* dst = ws + OFF_SEG + (((b * 8 + kt) * NSEG + g) * 2048);
    #pragma unroll
    for (int m = 0; m < 8; m++) dst[(i0 + 16 * m) * 16 + k] = acc[m];
}

// ---------------------------------------------------------------------------
// K3b: exclusive scan over the NSEG segments (in place)
// ---------------------------------------------------------------------------
__global__ void __launch_bounds__(256)
seg_scan(float* ws) {
    const int bk = blockIdx.x;  // b*8+kt
    const int tid = threadIdx.x;
    float* base = ws + OFF_SEG + (long long)bk * NSEG * 2048;
    for (int e = tid; e < 2048; e += 256) {
        float carry = 0.0f;
        for (int gseg = 0; gseg < NSEG; gseg++) {
            float* p = base + gseg * 2048 + e;
            float t = *p;
            *p = carry;
            carry += t;
        }
    }
}

// ---------------------------------------------------------------------------
// K3c: fused main — TDM double-buffered tiles + fp32 WMMA + quantized prefix
// ---------------------------------------------------------------------------
__device__ __forceinline__ void tdm_load_tile(const float* gp, unsigned lds_byte_addr) {
    // D# for a 2D tile: 128 rows x 16 elems, 4B data, row stride 128 elems.
    unsigned long long ga = (unsigned long long)(uintptr_t)gp;
    u32x4 g0;
    g0.x = 1u;                                            // count=1, load, user
    g0.y = lds_byte_addr;                                 // lds_addr
    g0.z = (unsigned)(ga & 0xFFFFFFFFu);                  // global_addr[31:0]
    g0.w = (unsigned)((ga >> 32) & 0x01FFFFFFu) | (2u << 30);  // [56:32] | type=2
    i32x8 g1;
    g1[0] = 0x00020000;          // workgroup_mask=0, data_size=2 (4B)
    g1[1] = (int)(128u << 16);   // tensor_dim0[15:0] = 128
    g1[2] = (int)(128u << 16);   // tensor_dim0 hi=0, tensor_dim1[15:0] = 128
    g1[3] = (int)(16u << 16);    // tensor_dim1 hi=0, tile_dim0 = 16
    g1[4] = 128;                 // tile_dim1 = 128, tile_dim2 = 0
    g1[5] = 128;                 // tensor_dim0_stride[31:0] = 128
    g1[6] = 0;                   // stride hi / tensor_dim1_stride lo
    g1[7] = 0;
    i32x4 z4 = {0, 0, 0, 0};
    i32x8 z8 = {0, 0, 0, 0, 0, 0, 0, 0};
    __builtin_amdgcn_tensor_load_to_lds(g0, g1, z4, z4, z8, 0);
}

__global__ void __launch_bounds__(256)
fused_main(const float* accum_in, float* ws, float* out_acc) {
    const int g = blockIdx.x, kt = blockIdx.y, b = blockIdx.z;
    const int tid = threadIdx.x;
    const int wv = tid >> 5;          // wave id = it tile 0..7
    const int lane = tid & 31;
    const int lm = lane & 15;
    const int hi = lane >> 4;         // 0 or 1
    const int it = wv;

    const float* tw = ws + OFF_TW;
    const float* ksp = ws + OFF_KS;
    const float* vsp = ws + OFF_VS;
    const float* qp  = ws + OFF_Q;
    float* mixed = ws + OFF_MIX;

    __shared__ __align__(16) float accT[2][Dm * 16];   // double-buffered TDM dst
    __shared__ float ks_s[Dm], q_s[Dm], vs_s[16], mix_s[16];

    // preload A fragments: tw rows [it*16, it*16+16), all 128 K split into 32 x4
    v2f afrag[32];
    const int klo = hi ? 2 : 0;
    {
        int arow = it * 16 + lm;
        #pragma unroll
        for (int jc = 0; jc < 32; jc++) {
            afrag[jc].x = tw[arow * Dm + jc * 4 + klo];
            afrag[jc].y = tw[arow * Dm + jc * 4 + klo + 1];
        }
    }

    // init running prefix tile from the segment carry (exclusive scan)
    v8f prefix;
    {
        const float* carry = ws + OFF_SEG + (((b * 8 + kt) * NSEG + g) * 2048);
        #pragma unroll
        for (int r = 0; r < 8; r++) {
            int M = r + hi * 8;
            prefix[r] = carry[(it * 16 + M) * 16 + lm];
        }
    }

    float maxabs = ws[OFF_MAX];
    float step = maxabs / QHALF;
    bool quant = maxabs > 1e-12f;
    float inv = quant ? 1.0f / step : 0.0f;

    // preload first tile via TDM (wave 0 issues, tracked on TENSORcnt)
    {
        const int t0 = g * SEGLEN;
        const float* gp = accum_in + (long long)(b * Tm + t0) * Dm * Dm + kt * 16;
        if (wv == 0) {
            tdm_load_tile(gp, (unsigned)(uintptr_t)(&accT[0][0]));
            __builtin_amdgcn_s_wait_tensorcnt(0);
        }
    }

    for (int tt = 0; tt < SEGLEN; tt++) {
        const int cur = tt & 1;
        const int t = g * SEGLEN + tt;
        const long long tokbase = (long long)(b * Tm + t) * Dm;
        const float* Ain = accum_in + tokbase * Dm + kt * 16;

        // issue TDM for next token into the other buffer; stage token vectors
        if (wv == 0 && tt + 1 < SEGLEN)
            tdm_load_tile(Ain + Dm * Dm, (unsigned)(uintptr_t)(&accT[cur ^ 1][0]));
        if (wv == 1 && tt + 4 < SEGLEN)
            __builtin_prefetch(Ain + 4 * Dm * Dm, 0, 1);   // warm GL2 ahead of TDM
        if (tid < Dm) {
            ks_s[tid] = ksp[tokbase + tid];
            q_s[tid]  = qp[tokbase + tid];
        } else if (tid < Dm + 16) {
            vs_s[tid - Dm] = vsp[tokbase + kt * 16 + (tid - Dm)];
        }
        if (tid < 16) mix_s[tid] = 0.0f;
        __syncthreads();   // staging + current TDM tile visible to all waves

        // D = tw_tile (16x128) @ Acol (128x16): 32 fp32 WMMAs, dual accumulators,
        // B-fragments staged 8 at a time for deep dscnt lookahead
        const float* at = &accT[cur][0];
        v8f c0 = {}, c1 = {};
        #pragma unroll
        for (int jb = 0; jb < 4; jb++) {
            v2f bf[8];
            #pragma unroll
            for (int u = 0; u < 8; u++) {
                int jc = jb * 8 + u;
                bf[u].x = at[(jc * 4 + klo) * 16 + lm];
                bf[u].y = at[(jc * 4 + klo + 1) * 16 + lm];
            }
            #pragma unroll
            for (int u = 0; u < 8; u++) {
                int jc = jb * 8 + u;
                if (u & 1)
                    c1 = __builtin_amdgcn_wmma_f32_16x16x4_f32(
                        false, afrag[jc], false, bf[u], (short)0, c1, false, false);
                else
                    c0 = __builtin_amdgcn_wmma_f32_16x16x4_f32(
                        false, afrag[jc], false, bf[u], (short)0, c0, false, false);
            }
        }
        v8f c = c0 + c1;

        // quantized rank-1 prefix update + write acc + partial q.acc
        float p = 0.0f;
        const int kglob = kt * 16 + lm;
        const float vv = vs_s[lm];
        float* orow = out_acc + tokbase * Dm;
        #pragma unroll
        for (int r = 0; r < 8; r++) {
            int M = r + hi * 8;
            int i = it * 16 + M;
            unsigned long long idx =
                (((unsigned long long)(b * Tm + t) * Dm + i) * Dm + kglob);
            prefix[r] += squant(ks_s[i] * vv, idx, step, inv, quant);
            float val = c[r] + prefix[r];
            orow[(long long)i * Dm + kglob] = val;
            p += q_s[i] * val;
        }
        atomicAdd(&mix_s[lm], p);  // ds_add_f32
        if (wv == 0) __builtin_amdgcn_s_wait_tensorcnt(0);  // next tile landed
        __syncthreads();
        if (tid < 16) mixed[tokbase + kt * 16 + tid] = mix_s[tid];
        __syncthreads();   // mix_s consumed before next iteration rezeroes
    }
}

// ---------------------------------------------------------------------------
// K4: out = state + mixed @ ow^T
// ---------------------------------------------------------------------------
__global__ void __launch_bounds__(128)
finalize(const float* state, const float* ws, float* out) {
    const int tok = blockIdx.x;
    const int j = threadIdx.x;
    const float* owT = ws + OFF_OWT;
    __shared__ float mr[Dm];
    mr[j] = ws[OFF_MIX + tok * Dm + j];
    __syncthreads();
    float a = 0.0f;
    for (int k2 = 0; k2 < Dm; k2++) a += mr[k2] * owT[k2 * Dm + j];
    out[tok * Dm + j] = state[tok * Dm + j] + a;
}

// ---------------------------------------------------------------------------
extern "C" void kernel_launch(void* const* d_in, const int* in_sizes, int n_in,
                              void* d_out, int out_size, void* d_ws, size_t ws_size,
                              hipStream_t stream) {
    const float* state = (const float*)d_in[0];
    const float* accum = (const float*)d_in[1];
    const float* qW = (const float*)d_in[2];
    const float* kW = (const float*)d_in[3];
    const float* vW = (const float*)d_in[4];
    const float* sW = (const float*)d_in[5];
    const float* oW = (const float*)d_in[6];
    const float* hl = (const float*)d_in[7];
    const float* hr = (const float*)d_in[8];

    float* ws = (float*)d_ws;
    float* out_main = (float*)d_out;                    // [B,T,D]
    float* out_acc  = (float*)d_out + (size_t)BT * Dm;  // [B,T,D,D]

    prep_weights<<<1, 256, 0, stream>>>(qW, kW, vW, sW, oW, hl, hr, ws);
    project_tokens<<<BT, 128, 0, stream>>>(state, ws);
    seg_outer<<<dim3(NSEG, 8, Bm), 256, 0, stream>>>(ws);
    seg_scan<<<Bm * 8, 256, 0, stream>>>(ws);
    fused_main<<<dim3(NSEG, 8, Bm), 256, 0, stream>>>(accum, ws, out_acc);
    finalize<<<BT, 128, 0, stream>>>(state, ws, out_main);
}